// GraphAttention_71081708749156
// MI455X (gfx1250) — compile-verified
//
#include <hip/hip_runtime.h>
#include <hip/hip_bf16.h>
#include <math.h>

typedef __attribute__((ext_vector_type(2))) float v2f_;
typedef __attribute__((ext_vector_type(8))) float v8f_;

#define DK  128       // DIN == DOUT == 128
#define LDA 132       // padded LDS row stride (floats): 132%64=4 -> conflict-free
#define MT_PER_BLOCK 8

// -------------------------------------------------------------------------
// Kernel 1: h = x @ W + b via V_WMMA_F32_16X16X4_F32.
// Block = 8 waves = 16 rows x 128 cols. A tile staged in LDS (shared by all
// 8 waves); each wave holds its entire 128x16 B tile in 32 v2f registers and
// sweeps MT_PER_BLOCK row tiles, so W and x are each read ~once from HBM.
// -------------------------------------------------------------------------
__global__ void gat_gemm_wmma(const float* __restrict__ x,
                              const float* __restrict__ W,
                              const float* __restrict__ bias,
                              float* __restrict__ h, int Nn) {
  __shared__ float As[16 * LDA];                 // 8448 B

  const int tid   = (int)threadIdx.x;
  const int wave  = tid >> 5;                    // 0..7 -> column tile
  const int lane  = tid & 31;
  const int m     = lane & 15;                   // row within tile (A) / col (B,C)
  const int khalf = lane >> 4;                   // 0: K=0,1   1: K=2,3
  const int col0  = wave << 4;

  // ---- preload B tile (128 x 16) into registers: 32 k-steps x v2f ----
  v2f_ breg[32];
  #pragma unroll
  for (int ks = 0; ks < 32; ++ks) {
    const int kk = (ks << 2) + (khalf << 1);
    v2f_ b;
    b.x = W[(size_t)kk       * DK + col0 + m];
    b.y = W[(size_t)(kk + 1) * DK + col0 + m];
    breg[ks] = b;
  }
  const float bv = bias[col0 + m];

  const int nTilesM = (Nn + 15) >> 4;
  const int tile0   = (int)blockIdx.x * MT_PER_BLOCK;
  const float4* __restrict__ x4 = (const float4*)x;

  for (int t = 0; t < MT_PER_BLOCK; ++t) {
    const int mt = tile0 + t;
    if (mt >= nTilesM) break;                    // uniform across block
    const int row0 = mt << 4;

    __syncthreads();                             // protect As reuse
    // ---- stage 16x128 A tile into padded LDS (512 float4 loads) ----
    for (int i = tid; i < 512; i += 256) {
      const int r  = i >> 5;                     // tile row
      const int wi = i & 31;                     // float4 within row
      int gr = row0 + r; if (gr >= Nn) gr = Nn - 1;
      const float4 v = x4[(size_t)gr * 32 + wi];
      float2* p = (float2*)(As + r * LDA + (wi << 2));
      p[0] = make_float2(v.x, v.y);
      p[1] = make_float2(v.z, v.w);
    }
    __syncthreads();

    // ---- 32 chained WMMAs over K=128 ----
    v8f_ c = {};
    #pragma unroll
    for (int ks = 0; ks < 32; ++ks) {
      const int kk = (ks << 2) + (khalf << 1);
      v2f_ a;
      a.x = As[m * LDA + kk];
      a.y = As[m * LDA + kk + 1];
      c = __builtin_amdgcn_wmma_f32_16x16x4_f32(false, a, false, breg[ks],
                                                (short)0, c, false, false);
    }

    // ---- epilogue: bias + store (fast path when full tile in range) ----
    const int rbase = row0 + (khalf << 3);       // C/D rows i / i+8
    if (row0 + 16 <= Nn) {
      #pragma unroll
      for (int i = 0; i < 8; ++i)
        h[(size_t)(rbase + i) * DK + col0 + m] = c[i] + bv;
    } else {
      #pragma unroll
      for (int i = 0; i < 8; ++i) {
        const int r = rbase + i;
        if (r < Nn) h[(size_t)r * DK + col0 + m] = c[i] + bv;
      }
    }
  }
}

// -------------------------------------------------------------------------
// Kernel 2: per-node projections  s_dst[n] = h[n]·a_w[:128],
//                                 s_src[n] = h[n]·a_w[128:]
// One wave per node; lane owns 4 channels (float4), wave shuffle reduce.
// -------------------------------------------------------------------------
__global__ void gat_proj(const float* __restrict__ h,
                         const float* __restrict__ a_w,
                         float* __restrict__ s_dst,
                         float* __restrict__ s_src, int Nn) {
  const int wid  = (int)((blockIdx.x * blockDim.x + threadIdx.x) >> 5);
  const int lane = (int)(threadIdx.x & 31);
  if (wid >= Nn) return;

  const float4 hv = ((const float4*)(h + (size_t)wid * DK))[lane];
  const float4 wd = ((const float4*)a_w)[lane];
  const float4 ws = ((const float4*)(a_w + DK))[lane];

  float pd = hv.x * wd.x + hv.y * wd.y + hv.z * wd.z + hv.w * wd.w;
  float ps = hv.x * ws.x + hv.y * ws.y + hv.z * ws.z + hv.w * ws.w;
  #pragma unroll
  for (int off = 16; off > 0; off >>= 1) {
    pd += __shfl_xor(pd, off, 32);
    ps += __shfl_xor(ps, off, 32);
  }
  if (lane == 0) { s_dst[wid] = pd; s_src[wid] = ps; }
}

// -------------------------------------------------------------------------
// Kernel 3: segment softmax + weighted aggregation, one wave per dst node.
// edge_dst is sorted -> binary search the segment; no atomics, deterministic.
// Lane owns output channels [4*lane, 4*lane+4); h gathers stay L2-resident.
// -------------------------------------------------------------------------
__device__ __forceinline__ int lower_bound_i32(const int* __restrict__ a,
                                               int n, int key) {
  int lo = 0, hi = n;
  while (lo < hi) {
    int mid = (lo + hi) >> 1;
    if (a[mid] < key) lo = mid + 1; else hi = mid;
  }
  return lo;
}

__global__ void gat_aggregate(const float* __restrict__ h,
                              const float* __restrict__ s_dst,
                              const float* __restrict__ s_src,
                              const float* __restrict__ a_b,
                              const int* __restrict__ edge_src,
                              const int* __restrict__ edge_dst,
                              int E, float* __restrict__ out, int Nn) {
  const int n    = (int)((blockIdx.x * blockDim.x + threadIdx.x) >> 5);
  const int lane = (int)(threadIdx.x & 31);
  if (n >= Nn) return;

  const int lo = lower_bound_i32(edge_dst, E, n);
  const int hi = lower_bound_i32(edge_dst, E, n + 1);

  float4* orow = (float4*)(out + (size_t)n * DK);
  if (lo == hi) {                       // no incoming edges -> zero row
    float4 z; z.x = z.y = z.z = z.w = 0.f;
    orow[lane] = z;
    return;
  }

  const float sdn = s_dst[n] + a_b[0];

  // ---- pass 1: segment max of leakyrelu(logit) ----
  float mloc = -INFINITY;
  for (int e = lo + lane; e < hi; e += 32) {
    float l = sdn + s_src[edge_src[e]];
    l = (l >= 0.f) ? l : 0.01f * l;
    mloc = fmaxf(mloc, l);
  }
  #pragma unroll
  for (int off = 16; off > 0; off >>= 1)
    mloc = fmaxf(mloc, __shfl_xor(mloc, off, 32));

  // ---- pass 2: accumulate denom and weighted feature sum ----
  float denom = 0.f;
  float4 acc; acc.x = acc.y = acc.z = acc.w = 0.f;

  for (int base = lo; base < hi; base += 32) {
    const int idx   = base + lane;
    const bool ok   = idx < hi;
    const int  msrc = ok ? edge_src[idx] : 0;
    float w = 0.f;
    if (ok) {
      float l = sdn + s_src[msrc];
      l = (l >= 0.f) ? l : 0.01f * l;
      w = __expf(l - mloc);
    }
    const int cnt = (hi - base) < 32 ? (hi - base) : 32;
    for (int j = 0; j < cnt; ++j) {
      const float wj  = __shfl(w, j, 32);
      const int   sj  = __shfl(msrc, j, 32);
      denom += wj;
      const float4 hv = ((const float4*)(h + (size_t)sj * DK))[lane];
      acc.x += wj * hv.x;
      acc.y += wj * hv.y;
      acc.z += wj * hv.z;
      acc.w += wj * hv.w;
    }
  }

  const float inv = 1.f / fmaxf(denom, 1e-16f);
  acc.x *= inv; acc.y *= inv; acc.z *= inv; acc.w *= inv;
  orow[lane] = acc;
}

// -------------------------------------------------------------------------
extern "C" void kernel_launch(void* const* d_in, const int* in_sizes, int n_in,
                              void* d_out, int out_size, void* d_ws, size_t ws_size,
                              hipStream_t stream) {
  const float* x     = (const float*)d_in[0];
  const int*   esrc  = (const int*)  d_in[1];
  const int*   edst  = (const int*)  d_in[2];
  const float* W     = (const float*)d_in[3];
  const float* bias  = (const float*)d_in[4];
  const float* a_w   = (const float*)d_in[5];
  const float* a_b   = (const float*)d_in[6];
  float*       out   = (float*)d_out;

  const int Nn = in_sizes[0] / DK;     // 50000
  const int E  = in_sizes[1];          // 800000

  char*  ws    = (char*)d_ws;
  float* h     = (float*)ws;                                   // N*128 f32
  float* s_dst = (float*)(ws + (size_t)Nn * DK * sizeof(float));
  float* s_src = s_dst + Nn;

  // GEMM: 1 block per MT_PER_BLOCK row tiles, 256 threads (8 waves)
  const int nTilesM    = (Nn + 15) >> 4;
  const int gemmBlocks = (nTilesM + MT_PER_BLOCK - 1) / MT_PER_BLOCK;
  gat_gemm_wmma<<<gemmBlocks, 256, 0, stream>>>(x, W, bias, h, Nn);

  // per-node kernels: 1 wave per node, 8 waves per block
  const int nodeBlocks = (Nn + 7) >> 3;
  gat_proj<<<nodeBlocks, 256, 0, stream>>>(h, a_w, s_dst, s_src, Nn);
  gat_aggregate<<<nodeBlocks, 256, 0, stream>>>(h, s_dst, s_src, a_b,
                                                esrc, edst, E, out, Nn);
}